// Encoder_30416958390352
// MI455X (gfx1250) — compile-verified
//
#include <hip/hip_runtime.h>
#include <cstdint>

// ---------------------------------------------------------------------------
// Problem constants (from reference): N=50000, IN=64, L=128, STEPS=3, T=4, E=800000
// ---------------------------------------------------------------------------
constexpr int kN     = 50000;
constexpr int kIN    = 64;
constexpr int kL     = 128;
constexpr int kT     = 4;
constexpr int kSTEPS = 3;
constexpr int kE     = 800000;

typedef float v2f __attribute__((ext_vector_type(2)));
typedef float v8f __attribute__((ext_vector_type(8)));

// D = A(16x4, f32) * B(4x16, f32) + C(16x16, f32)  -- CDNA5 V_WMMA_F32_16X16X4_F32
static __device__ __forceinline__ v8f wmma4(v2f a, v2f b, v8f c) {
    return __builtin_amdgcn_wmma_f32_16x16x4_f32(false, a, false, b, (short)0, c,
                                                 false, false);
}

// Load a 2-element fragment (K-pair) for A (row-major activations, row=node)
// or B (row-major weights, row=output-col; B_wmma[k][n] = W[n][k]).
// lanes 0-15: K = {kb, kb+1}; lanes 16-31: K = {kb+2, kb+3}.
static __device__ __forceinline__ v2f ldfrag(const float* __restrict__ base,
                                             int rowStride, int row, int kb, int half) {
    const float2* p = reinterpret_cast<const float2*>(base + (size_t)row * rowStride);
    float2 t = p[(kb >> 1) + half];
    v2f r; r[0] = t.x; r[1] = t.y;
    return r;
}

// ---------------------------------------------------------------------------
// bond[e] = argmax(edge_attr[e, 0:4])  (first-max, matching jnp.argmax)
// ---------------------------------------------------------------------------
__global__ void bond_kernel(const float* __restrict__ ea, int* __restrict__ bond) {
    int e = blockIdx.x * blockDim.x + threadIdx.x;
    if (e >= kE) return;
    float4 a = reinterpret_cast<const float4*>(ea)[e];
    int best = 0; float bv = a.x;
    if (a.y > bv) { bv = a.y; best = 1; }
    if (a.z > bv) { bv = a.z; best = 2; }
    if (a.w > bv) { bv = a.w; best = 3; }
    bond[e] = best;
}

// ---------------------------------------------------------------------------
// x0 = relu(x @ lin_w^T + lin_b); also written to m (initial message state)
// A: [N,64], W: [128,64]   one 16x16 tile per wave, K=64
// ---------------------------------------------------------------------------
__global__ __launch_bounds__(128) void encoder_kernel(
        const float* __restrict__ x, const float* __restrict__ lw,
        const float* __restrict__ lb, float* __restrict__ x0, float* __restrict__ m) {
    const int lane = threadIdx.x & 31, wv = threadIdx.x >> 5;
    const int l16 = lane & 15, half = lane >> 4;
    const int mb = blockIdx.x * 16;
    const int cb = (blockIdx.y * 4 + wv) * 16;
    const int arow = mb + l16, brow = cb + l16;
    v8f acc = {};
    for (int kb = 0; kb < kIN; kb += 4) {
        v2f a = ldfrag(x,  kIN, arow, kb, half);
        v2f b = ldfrag(lw, kIN, brow, kb, half);
        acc = wmma4(a, b, acc);
    }
    const int col = cb + l16;
    const float bias = lb[col];
#pragma unroll
    for (int r = 0; r < 8; ++r) {
        int row = mb + r + 8 * half;
        float v = fmaxf(acc[r] + bias, 0.0f);
        x0[(size_t)row * kL + col] = v;
        m [(size_t)row * kL + col] = v;
    }
}

// ---------------------------------------------------------------------------
// Ht = m @ W^T   (W = gnn_w[step][t], [128,128]); K=128, no epilogue
// ---------------------------------------------------------------------------
__global__ __launch_bounds__(128) void msg_gemm_kernel(
        const float* __restrict__ m, const float* __restrict__ W,
        float* __restrict__ Ht) {
    const int lane = threadIdx.x & 31, wv = threadIdx.x >> 5;
    const int l16 = lane & 15, half = lane >> 4;
    const int mb = blockIdx.x * 16;
    const int cb = (blockIdx.y * 4 + wv) * 16;
    const int arow = mb + l16, brow = cb + l16;
    v8f acc = {};
    for (int kb = 0; kb < kL; kb += 4) {
        v2f a = ldfrag(m, kL, arow, kb, half);
        v2f b = ldfrag(W, kL, brow, kb, half);
        acc = wmma4(a, b, acc);
    }
    const int col = cb + l16;
#pragma unroll
    for (int r = 0; r < 8; ++r) {
        int row = mb + r + 8 * half;
        Ht[(size_t)row * kL + col] = acc[r];
    }
}

// ---------------------------------------------------------------------------
// Scatter-add: for edges of type `type`:  agg[dst] += 2 * Ht[src]
// (factor 2 accounts for the reference's tile(edge_index,(1,2)) duplication)
// One wave handles one edge: 128 floats = 32 lanes x float4.
// ---------------------------------------------------------------------------
__global__ __launch_bounds__(256) void scatter_kernel(
        const float* __restrict__ Ht, const long long* __restrict__ ei,
        const int* __restrict__ bond, float* __restrict__ agg, int type) {
    const int lane = threadIdx.x & 31;
    const int gw = blockIdx.x * (blockDim.x >> 5) + (threadIdx.x >> 5);
    const int nw = gridDim.x * (blockDim.x >> 5);
    for (int e = gw; e < kE; e += nw) {
        if (bond[e] != type) continue;            // uniform per wave
        int src = (int)ei[e];
        int dst = (int)ei[kE + e];
        float4 v = reinterpret_cast<const float4*>(Ht + (size_t)src * kL)[lane];
        float* ar = agg + (size_t)dst * kL + lane * 4;
        atomicAdd(ar + 0, 2.0f * v.x);
        atomicAdd(ar + 1, 2.0f * v.y);
        atomicAdd(ar + 2, 2.0f * v.z);
        atomicAdd(ar + 3, 2.0f * v.w);
    }
}

// ---------------------------------------------------------------------------
// Fused GRU cell + relu:
//   gi = agg @ w_ih^T + b_ih ; gh = x0 @ w_hh^T + b_hh ; h = x0
//   r = sig(i_r+h_r); z = sig(i_z+h_z); n = tanh(i_n + r*h_n)
//   m = relu((1-z)*n + z*h)
// 6 WMMA accumulator tiles per wave (3 gates x {ih, hh}), K=128.
// ---------------------------------------------------------------------------
__global__ __launch_bounds__(128) void gru_kernel(
        const float* __restrict__ agg, const float* __restrict__ x0,
        const float* __restrict__ w_ih, const float* __restrict__ w_hh,
        const float* __restrict__ b_ih, const float* __restrict__ b_hh,
        float* __restrict__ m) {
    const int lane = threadIdx.x & 31, wv = threadIdx.x >> 5;
    const int l16 = lane & 15, half = lane >> 4;
    const int mb = blockIdx.x * 16;
    const int cb = (blockIdx.y * 4 + wv) * 16;
    const int arow = mb + l16;
    const int cr = cb + l16, cz = kL + cb + l16, cn = 2 * kL + cb + l16;
    v8f air = {}, aiz = {}, ain = {}, ahr = {}, ahz = {}, ahn = {};
    for (int kb = 0; kb < kL; kb += 4) {
        v2f a1 = ldfrag(agg, kL, arow, kb, half);
        v2f a2 = ldfrag(x0,  kL, arow, kb, half);
        v2f b;
        b = ldfrag(w_ih, kL, cr, kb, half); air = wmma4(a1, b, air);
        b = ldfrag(w_ih, kL, cz, kb, half); aiz = wmma4(a1, b, aiz);
        b = ldfrag(w_ih, kL, cn, kb, half); ain = wmma4(a1, b, ain);
        b = ldfrag(w_hh, kL, cr, kb, half); ahr = wmma4(a2, b, ahr);
        b = ldfrag(w_hh, kL, cz, kb, half); ahz = wmma4(a2, b, ahz);
        b = ldfrag(w_hh, kL, cn, kb, half); ahn = wmma4(a2, b, ahn);
    }
    const int col = cb + l16;
    const float bir = b_ih[col], biz = b_ih[kL + col], bin2 = b_ih[2 * kL + col];
    const float bhr = b_hh[col], bhz = b_hh[kL + col], bhn2 = b_hh[2 * kL + col];
#pragma unroll
    for (int r = 0; r < 8; ++r) {
        int row = mb + r + 8 * half;
        float ir = air[r] + bir, iz = aiz[r] + biz, in2 = ain[r] + bin2;
        float hr = ahr[r] + bhr, hz = ahz[r] + bhz, hn = ahn[r] + bhn2;
        float rg = 1.0f / (1.0f + expf(-(ir + hr)));
        float zg = 1.0f / (1.0f + expf(-(iz + hz)));
        float ng = tanhf(in2 + rg * hn);
        float h  = x0[(size_t)row * kL + col];
        float o  = (1.0f - zg) * ng + zg * h;
        m[(size_t)row * kL + col] = fmaxf(o, 0.0f);
    }
}

// ---------------------------------------------------------------------------
// mu = m @ mu_w^T + mu_b ; logvar = m @ lv_w^T + lv_b  (shared A-fragments)
// out layout: [mu (N*L) | logvar (N*L)]
// ---------------------------------------------------------------------------
__global__ __launch_bounds__(128) void head_kernel(
        const float* __restrict__ m,
        const float* __restrict__ mu_w, const float* __restrict__ mu_b,
        const float* __restrict__ lv_w, const float* __restrict__ lv_b,
        float* __restrict__ out) {
    const int lane = threadIdx.x & 31, wv = threadIdx.x >> 5;
    const int l16 = lane & 15, half = lane >> 4;
    const int mb = blockIdx.x * 16;
    const int cb = (blockIdx.y * 4 + wv) * 16;
    const int arow = mb + l16, brow = cb + l16;
    v8f amu = {}, alv = {};
    for (int kb = 0; kb < kL; kb += 4) {
        v2f a  = ldfrag(m,    kL, arow, kb, half);
        v2f b1 = ldfrag(mu_w, kL, brow, kb, half);
        v2f b2 = ldfrag(lv_w, kL, brow, kb, half);
        amu = wmma4(a, b1, amu);
        alv = wmma4(a, b2, alv);
    }
    const int col = cb + l16;
    const float bm = mu_b[col], bl = lv_b[col];
    const size_t half2 = (size_t)kN * kL;
#pragma unroll
    for (int r = 0; r < 8; ++r) {
        int row = mb + r + 8 * half;
        out[(size_t)row * kL + col]         = amu[r] + bm;
        out[half2 + (size_t)row * kL + col] = alv[r] + bl;
    }
}

// ---------------------------------------------------------------------------
// Host-side launcher
// Workspace layout (floats): x0 | m | agg | Ht | bond(int)  ~= 105.7 MB
// ---------------------------------------------------------------------------
extern "C" void kernel_launch(void* const* d_in, const int* in_sizes, int n_in,
                              void* d_out, int out_size, void* d_ws, size_t ws_size,
                              hipStream_t stream) {
    (void)in_sizes; (void)n_in; (void)out_size; (void)ws_size;

    const float*     x     = (const float*)d_in[0];
    const long long* ei    = (const long long*)d_in[1];   // int64 [2, E]
    const float*     ea    = (const float*)d_in[2];
    const float*     lin_w = (const float*)d_in[3];
    const float*     lin_b = (const float*)d_in[4];
    const float*     gnn_w = (const float*)d_in[5];
    const float*     w_ih  = (const float*)d_in[6];
    const float*     w_hh  = (const float*)d_in[7];
    const float*     b_ih  = (const float*)d_in[8];
    const float*     b_hh  = (const float*)d_in[9];
    const float*     mu_w  = (const float*)d_in[10];
    const float*     mu_b  = (const float*)d_in[11];
    const float*     lv_w  = (const float*)d_in[12];
    const float*     lv_b  = (const float*)d_in[13];
    float* out = (float*)d_out;

    const size_t NL = (size_t)kN * kL;
    float* x0  = (float*)d_ws;
    float* m   = x0 + NL;
    float* agg = m + NL;
    float* Ht  = agg + NL;
    int*   bond = (int*)(Ht + NL);

    dim3 gemmGrid(kN / 16, 2);   // 3125 row-tiles x (2 blocks * 4 waves = 8 col-tiles)
    dim3 gemmBlk(128);

    bond_kernel<<<(kE + 255) / 256, 256, 0, stream>>>(ea, bond);
    encoder_kernel<<<gemmGrid, gemmBlk, 0, stream>>>(x, lin_w, lin_b, x0, m);

    for (int s = 0; s < kSTEPS; ++s) {
        hipMemsetAsync(agg, 0, NL * sizeof(float), stream);
        for (int t = 0; t < kT; ++t) {
            const float* W = gnn_w + (size_t)(s * kT + t) * kL * kL;
            msg_gemm_kernel<<<gemmGrid, gemmBlk, 0, stream>>>(m, W, Ht);
            scatter_kernel<<<4096, 256, 0, stream>>>(Ht, ei, bond, agg, t);
        }
        gru_kernel<<<gemmGrid, gemmBlk, 0, stream>>>(agg, x0, w_ih, w_hh, b_ih, b_hh, m);
    }
    head_kernel<<<gemmGrid, gemmBlk, 0, stream>>>(m, mu_w, mu_b, lv_w, lv_b, out);
}